// Conv1Layer_73065983639637
// MI455X (gfx1250) — compile-verified
//
#include <hip/hip_runtime.h>

typedef float v2f __attribute__((ext_vector_type(2)));
typedef float v8f __attribute__((ext_vector_type(8)));

#define LAT 256
#define LATLAT (LAT * LAT)

// ---------------------------------------------------------------------------
// Kernel 1: the dominant cost. 134 MB of zeros -> coalesced b128 stores,
// grid-stride. Pure store-bandwidth bound (~5.8 us at 23.3 TB/s; output even
// fits in the 192 MB L2).
// ---------------------------------------------------------------------------
__global__ void zero_fill_kernel(float* __restrict__ out, int total) {
  const int count4 = total >> 2;
  float4* __restrict__ o4 = (float4*)out;
  const int idx = blockIdx.x * blockDim.x + threadIdx.x;
  const int stride = gridDim.x * blockDim.x;
  const float4 z = make_float4(0.f, 0.f, 0.f, 0.f);
  for (int i = idx; i < count4; i += stride) o4[i] = z;
  if (idx == 0) {
    for (int t = count4 << 2; t < total; ++t) out[t] = 0.f;
  }
}

// A-matrix element: rows 0..7 = v0 * J8 (anti-identity), rows 8..15 = -J8,
// columns >= 8 are zero (K is effectively 8).
__device__ __forceinline__ float Aval(int m, int k, float v0) {
  if (k >= 8) return 0.f;
  if (m < 8) return (m + k == 7) ? v0 : 0.f;
  return ((m - 8) + k == 7) ? -1.f : 0.f;
}

// ---------------------------------------------------------------------------
// Kernel 2: per-sample stamp. One wave32 per sample. Two chained
// v_wmma_f32_16x16x4_f32 compute T = A x (W*J8):
//   rows 0..7  = v0 * rot180(W)   (delta at (0,0), v0=0 if deltas coincide)
//   rows 8..15 = -1 * rot180(W)   (delta at (y,x))
// f32 WMMA with a scaled-permutation A => bit-exact. Lanes then scatter the
// 8x8 stamps with wrap-around via global_atomic_add_f32 (handles overlap of
// the two footprints correctly).
// ---------------------------------------------------------------------------
__global__ void __launch_bounds__(32)
stamp_kernel(const float* __restrict__ weight,
             const int* __restrict__ x_seps,
             const int* __restrict__ y_seps,
             float* __restrict__ out) {
  const int n    = blockIdx.x;
  const int lane = threadIdx.x;   // 0..31, wave32
  const int half = lane >> 4;     // lane group: 0 or 1
  const int m    = lane & 15;     // A row index / B,C column index

  const int x = x_seps[n];
  const int y = y_seps[n];
  // Overwrite semantics: the -1 assignment clobbers the +1 when (y,x)==(0,0).
  const float v0 = (x == 0 && y == 0) ? 0.f : 1.f;

  // Fragment layout rule (ISA 7.12.2, 32-bit A/B 16x4 resp 4x16):
  //   chunk k0, VGPR g, lane-half h  ->  K = 4*k0 + 2*h + g
  // B = W * J8 padded to 16x16: B[k][col] = W[k][7-col] (k<8, col<8).
  v2f a0, a1, b0, b1;
  {
    const int k = 2 * half;                       // chunk 0: K in {0,1,2,3}
    a0.x = Aval(m, k, v0);
    a0.y = Aval(m, k + 1, v0);
    b0.x = (m < 8) ? weight[(k)     * 8 + (7 - m)] : 0.f;
    b0.y = (m < 8) ? weight[(k + 1) * 8 + (7 - m)] : 0.f;
  }
  {
    const int k = 4 + 2 * half;                   // chunk 1: K in {4,5,6,7}
    a1.x = Aval(m, k, v0);
    a1.y = Aval(m, k + 1, v0);
    b1.x = (m < 8) ? weight[(k)     * 8 + (7 - m)] : 0.f;
    b1.y = (m < 8) ? weight[(k + 1) * 8 + (7 - m)] : 0.f;
  }

  // EXEC is all-ones here (no divergent branch yet) as WMMA requires.
  v8f c = {};
  c = __builtin_amdgcn_wmma_f32_16x16x4_f32(
      /*neg_a=*/false, a0, /*neg_b=*/false, b0,
      /*c_mod=*/(short)0, c, /*reuse_a=*/false, /*reuse_b=*/false);
  c = __builtin_amdgcn_wmma_f32_16x16x4_f32(
      /*neg_a=*/false, a1, /*neg_b=*/false, b1,
      /*c_mod=*/(short)0, c, /*reuse_a=*/false, /*reuse_b=*/false);
  // C layout: c[g] = T[g + 8*half][m]
  //   half==0 lanes: rows 0..7  -> stamp for delta at (0,0), scale v0
  //   half==1 lanes: rows 8..15 -> stamp for delta at (y,x), scale -1

  if (m < 8) {
    const int dy = half ? y : 0;
    const int dx = half ? x : 0;
    const int jj = (dx - 4 + m) & (LAT - 1);
    float* __restrict__ base = out + (size_t)n * LATLAT;
#pragma unroll
    for (int a = 0; a < 8; ++a) {
      const int ii = (dy - 4 + a) & (LAT - 1);
      atomicAdd(base + ii * LAT + jj, c[a]);
    }
  }
}

extern "C" void kernel_launch(void* const* d_in, const int* in_sizes, int n_in,
                              void* d_out, int out_size, void* d_ws, size_t ws_size,
                              hipStream_t stream) {
  // setup_inputs order: temps (f32, unused), x_seps (int), y_seps (int),
  //                     weight (f32, 64)
  const int*   x_seps = (const int*)d_in[1];
  const int*   y_seps = (const int*)d_in[2];
  const float* weight = (const float*)d_in[3];
  float*       out    = (float*)d_out;
  const int N = in_sizes[1];  // 512

  const int threads = 256;
  int blocks = ((out_size >> 2) + threads - 1) / threads;
  if (blocks > 16384) blocks = 16384;
  if (blocks < 1) blocks = 1;
  zero_fill_kernel<<<blocks, threads, 0, stream>>>(out, out_size);

  stamp_kernel<<<N, 32, 0, stream>>>(weight, x_seps, y_seps, out);
}